// GMSA_59133109731760
// MI455X (gfx1250) — compile-verified
//
#include <hip/hip_runtime.h>
#include <hip/hip_bf16.h>

// ---------------------------------------------------------------------------
// GMSA block for MI455X (gfx1250, wave32, WMMA, async-to-LDS).
// Pipeline: fold BN -> GEMM1 (384x192 @ 192xHW, wmma f32 16x16x4)
//           -> 3x shifted-window attention branches (ws=2,4,8)
//           -> GEMM2 (192x192 @ 192xHW, wmma f32 16x16x4)
// GEMM staging uses GLOBAL_LOAD_ASYNC_TO_LDS_* (ASYNCcnt-tracked DMA) with
// double buffering, overlapping the HBM fetch with the WMMA stream.
// ---------------------------------------------------------------------------

typedef float v2f __attribute__((ext_vector_type(2)));
typedef float v8f __attribute__((ext_vector_type(8)));
typedef int   gmsa_v4i __attribute__((__vector_size__(4 * sizeof(int))));

#define BATCH   8
#define CIN     192
#define C2      384
#define HH      192
#define WW2     192
#define HWPIX   (HH * WW2)   // 36864

#if defined(__AMDGCN__) && \
    __has_builtin(__builtin_amdgcn_global_load_async_to_lds_b32) && \
    __has_builtin(__builtin_amdgcn_global_load_async_to_lds_b128) && \
    __has_builtin(__builtin_amdgcn_s_wait_asynccnt)
#define GMSA_USE_ASYNC 1
#pragma message("GMSA: gfx1250 async-to-LDS path ENABLED")
#else
#pragma message("GMSA: async-to-LDS path DISABLED (builtins missing) - sync fallback")
#endif

// Global side: flat VA == AS1 VA, so an integer reinterpret is value-preserving.
// LDS side: per ISA aperture rules a flat LDS address maps to LDS by dropping
// the upper bits, so truncating to the low 32 bits yields the AS3 byte offset.
#define GMSA_GLB_I32(p)  ((__attribute__((address_space(1))) int*)(uintptr_t)(p))
#define GMSA_LDS_I32(p)  ((__attribute__((address_space(3))) int*)(unsigned)(uintptr_t)(p))
#define GMSA_GLB_V4I(p)  ((__attribute__((address_space(1))) gmsa_v4i*)(uintptr_t)(p))
#define GMSA_LDS_V4I(p)  ((__attribute__((address_space(3))) gmsa_v4i*)(unsigned)(uintptr_t)(p))

// ---------------------------------------------------------------------------
// Kernel 0: fold BatchNorm (inference) into the input-projection weights.
// ---------------------------------------------------------------------------
__global__ __launch_bounds__(256)
void fold_bn_kernel(const float* __restrict__ w_in, const float* __restrict__ b_in,
                    const float* __restrict__ gamma, const float* __restrict__ beta,
                    const float* __restrict__ mean, const float* __restrict__ var,
                    float* __restrict__ w_eff, float* __restrict__ b_eff)
{
    int idx = blockIdx.x * blockDim.x + threadIdx.x;
    if (idx < C2 * CIN) {
        int o = idx / CIN;
        float inv = gamma[o] * rsqrtf(var[o] + 1e-5f);
        w_eff[idx] = w_in[idx] * inv;
    }
    if (idx < C2) {
        float inv = gamma[idx] * rsqrtf(var[idx] + 1e-5f);
        b_eff[idx] = b_in[idx] * inv + beta[idx] - mean[idx] * inv;
    }
}

// ---------------------------------------------------------------------------
// Tiled GEMM with bias:  Y_b(MxN) = A(MxK) * X_b(KxN) + bias
// One wave computes a 16 x BN strip via v_wmma_f32_16x16x4_f32.
// Double-buffered LDS tiles; pad stride KB+4 keeps b64 fragment loads 8B
// aligned while spreading lanes across 16 bank-pairs (<=2-way conflicts).
//
// WMMA f32 16x16x4 register layouts (ISA 7.12.2):
//   A (16x4):  vgpr0 = K+0 (lanes 0-15, M=lane) / K+2 (lanes 16-31)
//              vgpr1 = K+1 / K+3            -> per-lane b64 LDS load, K-contig
//   B (4x16):  mirrored: vgpr0 = K+0 / K+2 with N = lane&15
//   C/D 16x16: vgpr r -> M = r (lanes 0-15) / r+8 (lanes 16-31), N = lane&15
// ---------------------------------------------------------------------------
template<int BM, int BN, int KB>
__global__ __launch_bounds__((BM / 16) * 32)
void gemm_bias_kernel(const float* __restrict__ A, const float* __restrict__ bias,
                      const float* __restrict__ X, float* __restrict__ Y,
                      int M, int K, int N,
                      long long xBatchStride, long long yBatchStride)
{
    constexpr int WAVES = BM / 16;
    constexpr int NT = BN / 16;
    constexpr int NTHREADS = WAVES * 32;
    constexpr int AST = KB + 4;   // padded LDS strides (floats)
    constexpr int BST = KB + 4;

    __shared__ float As[2][BM * AST];
    __shared__ float Bs[2][BN * BST];

    const int tid  = threadIdx.x;
    const int lane = tid & 31;
    const int wv   = tid >> 5;
    const int n0   = blockIdx.x * BN;
    const int m0   = blockIdx.y * BM;

    const float* __restrict__ Xb = X + (long long)blockIdx.z * xBatchStride;
    float* __restrict__ Yb       = Y + (long long)blockIdx.z * yBatchStride;

    v8f acc[NT];
    v8f vzero = {};
#pragma unroll
    for (int i = 0; i < NT; ++i) acc[i] = vzero;

    const int mrow = wv * 16 + (lane & 15);
    const int ksel = (lane >> 4) << 1;      // 0 for lanes 0-15, 2 for lanes 16-31

    // stage one KB-deep tile (A: b128 chunks, X: b32 transposed to [n][k])
    auto stage = [&](int buf, int kt) {
#pragma unroll
        for (int idx = tid; idx < BM * (KB / 4); idx += NTHREADS) {
            int m  = idx / (KB / 4);
            int kq = (idx % (KB / 4)) * 4;
            const float* src = &A[(long long)(m0 + m) * K + kt + kq];
            float* dst = &As[buf][m * AST + kq];
#if defined(GMSA_USE_ASYNC)
            __builtin_amdgcn_global_load_async_to_lds_b128(GMSA_GLB_V4I(src), GMSA_LDS_V4I(dst), 0, 0);
#else
            *(float4*)dst = *(const float4*)src;
#endif
        }
#pragma unroll
        for (int idx = tid; idx < BN * KB; idx += NTHREADS) {
            int k = idx / BN, n = idx % BN;   // global-contiguous along n
            const float* src = &Xb[(long long)(kt + k) * N + n0 + n];
            float* dst = &Bs[buf][n * BST + k];
#if defined(GMSA_USE_ASYNC)
            __builtin_amdgcn_global_load_async_to_lds_b32(GMSA_GLB_I32(src), GMSA_LDS_I32(dst), 0, 0);
#else
            *dst = *src;
#endif
        }
    };

    stage(0, 0);
#if defined(GMSA_USE_ASYNC)
    __builtin_amdgcn_s_wait_asynccnt(0);
#endif
    __syncthreads();

    const int T = K / KB;
    for (int t = 0; t < T; ++t) {
        const int buf = t & 1;
        if (t + 1 < T) {
            // prefetch next tile into the other buffer while computing this one
            stage(buf ^ 1, (t + 1) * KB);
        }

#pragma unroll
        for (int kk = 0; kk < KB; kk += 4) {
            v2f a = *(const v2f*)&As[buf][mrow * AST + kk + ksel];
#pragma unroll
            for (int nt = 0; nt < NT; ++nt) {
                int ncol = nt * 16 + (lane & 15);
                v2f b = *(const v2f*)&Bs[buf][ncol * BST + kk + ksel];
                acc[nt] = __builtin_amdgcn_wmma_f32_16x16x4_f32(
                    /*neg_a=*/false, a, /*neg_b=*/false, b,
                    /*c_mod=*/(short)0, acc[nt],
                    /*reuse_a=*/false, /*reuse_b=*/false);
            }
        }

        if (t + 1 < T) {
#if defined(GMSA_USE_ASYNC)
            __builtin_amdgcn_s_wait_asynccnt(0);
#endif
            __syncthreads();
        }
    }

    // --- epilogue: D layout -> global, add bias ---
#pragma unroll
    for (int nt = 0; nt < NT; ++nt) {
        int ncol = n0 + nt * 16 + (lane & 15);
#pragma unroll
        for (int r = 0; r < 8; ++r) {
            int m = m0 + wv * 16 + r + ((lane >> 4) << 3);
            Yb[(long long)m * N + ncol] = acc[nt][r] + bias[m];
        }
    }
}

// ---------------------------------------------------------------------------
// Shifted-window self-attention branch.
// Each 512-thread block handles a slab of 64 window-positions for all 8 heads
// (1 window for ws=8, 4 for ws=4, 16 for ws=2). Thread = one (position, head)
// score row. The two jnp.rolls cancel: read and write both use the original
// pixel ( (win*ws + d + sh) mod 192 ).
//   y channels (per branch slice of 128): [qv(2), head(8), c(8)]
//   z channel: branch*64 + head*8 + c
// ---------------------------------------------------------------------------
template<int WSZ>
__global__ __launch_bounds__(512)
void attn_kernel(const float* __restrict__ y, float* __restrict__ z, int branch)
{
    constexpr int SH   = WSZ / 2;
    constexpr int WIN2 = WSZ * WSZ;
    constexpr int NWB  = 64 / WIN2;        // windows per block
    constexpr int NW   = HH / WSZ;         // windows per image row

    const int tid  = threadIdx.x;
    const int p    = tid & 63;             // slab position
    const int head = tid >> 6;             // 0..7
    const int b    = blockIdx.y;

    const int wlocal = p / WIN2;
    const int r      = p % WIN2;
    const int dh     = r / WSZ;
    const int dw     = r % WSZ;
    const int wglob  = blockIdx.x * NWB + wlocal;
    const int wh     = wglob / NW;
    const int ww     = wglob % NW;
    const int oh     = (wh * WSZ + dh + SH) % HH;
    const int ow     = (ww * WSZ + dw + SH) % WW2;
    const int pix    = oh * WW2 + ow;
    const int co     = branch * 128;       // branch channel offset inside y

    __shared__ float qs[8][64][8];
    __shared__ float vs[8][64][8];

    const float* __restrict__ yb = y + (long long)b * C2 * HWPIX;
#pragma unroll
    for (int c = 0; c < 8; ++c)
        qs[head][p][c] = yb[(long long)(co + head * 8 + c) * HWPIX + pix];
#pragma unroll
    for (int c = 0; c < 8; ++c)
        vs[head][p][c] = yb[(long long)(co + 64 + head * 8 + c) * HWPIX + pix];
    __syncthreads();

    float q[8];
#pragma unroll
    for (int c = 0; c < 8; ++c) q[c] = qs[head][p][c];

    const int jbase = wlocal * WIN2;
    float s[WIN2];
    float mx = -3.402823466e+38f;
#pragma unroll 4
    for (int j = 0; j < WIN2; ++j) {
        float d = 0.f;
#pragma unroll
        for (int c = 0; c < 8; ++c) d += q[c] * qs[head][jbase + j][c];
        s[j] = d;
        mx = fmaxf(mx, d);
    }
    float sum = 0.f;
#pragma unroll 4
    for (int j = 0; j < WIN2; ++j) {
        s[j] = __expf(s[j] - mx);
        sum += s[j];
    }
    const float inv = 1.0f / sum;

    float o[8];
#pragma unroll
    for (int c = 0; c < 8; ++c) o[c] = 0.f;
#pragma unroll 4
    for (int j = 0; j < WIN2; ++j) {
#pragma unroll
        for (int c = 0; c < 8; ++c) o[c] += s[j] * vs[head][jbase + j][c];
    }

    float* __restrict__ zb = z + (long long)b * CIN * HWPIX;
#pragma unroll
    for (int c = 0; c < 8; ++c)
        zb[(long long)(branch * 64 + head * 8 + c) * HWPIX + pix] = o[c] * inv;
}

// ---------------------------------------------------------------------------
// Launch
// ---------------------------------------------------------------------------
extern "C" void kernel_launch(void* const* d_in, const int* in_sizes, int n_in,
                              void* d_out, int out_size, void* d_ws, size_t ws_size,
                              hipStream_t stream)
{
    const float* x     = (const float*)d_in[0];
    const float* w_in  = (const float*)d_in[1];
    const float* b_in  = (const float*)d_in[2];
    const float* gamma = (const float*)d_in[3];
    const float* beta  = (const float*)d_in[4];
    const float* mean  = (const float*)d_in[5];
    const float* var   = (const float*)d_in[6];
    const float* w_out = (const float*)d_in[7];
    const float* b_out = (const float*)d_in[8];
    float* out = (float*)d_out;

    // workspace carve-up (floats)
    float* wsf   = (float*)d_ws;
    float* w_eff = wsf;                                   // 384*192   = 73728
    float* b_eff = wsf + 73728;                           // 384
    float* y     = wsf + 74112;                           // 8*384*HW  = 113246208
    float* z     = y + (long long)BATCH * C2 * HWPIX;     // 8*192*HW  = 56623104

    // 0) fold BN into projection weights
    fold_bn_kernel<<<(C2 * CIN + 255) / 256, 256, 0, stream>>>(
        w_in, b_in, gamma, beta, mean, var, w_eff, b_eff);

    // 1) input projection GEMM: y = w_eff(384x192) @ x_b(192x36864) + b_eff
    {
        dim3 grid(HWPIX / 64, C2 / 128, BATCH);
        gemm_bias_kernel<128, 64, 16><<<grid, 256, 0, stream>>>(
            w_eff, b_eff, x, y, C2, CIN, HWPIX,
            (long long)CIN * HWPIX, (long long)C2 * HWPIX);
    }

    // 2) three shifted-window attention branches (ws = 2, 4, 8)
    {
        dim3 grid(576, BATCH);  // 9216/16 = 2304/4 = 576/1 = 576 slabs/batch
        attn_kernel<2><<<grid, 512, 0, stream>>>(y, z, 0);
        attn_kernel<4><<<grid, 512, 0, stream>>>(y, z, 1);
        attn_kernel<8><<<grid, 512, 0, stream>>>(y, z, 2);
    }

    // 3) output projection GEMM: out = w_out(192x192) @ z_b(192x36864) + b_out
    {
        dim3 grid(HWPIX / 64, CIN / 64, BATCH);
        gemm_bias_kernel<64, 64, 16><<<grid, 128, 0, stream>>>(
            w_out, b_out, z, out, CIN, CIN, HWPIX,
            (long long)CIN * HWPIX, (long long)CIN * HWPIX);
    }
}